// MyGCN_32727650795984
// MI455X (gfx1250) — compile-verified
//
#include <hip/hip_runtime.h>
#include <hip/hip_bf16.h>
#include <math.h>

typedef __attribute__((ext_vector_type(2))) float v2f;
typedef __attribute__((ext_vector_type(8))) float v8f;

#define GN_NODES 50000
#define GIN_C    96
#define GHID_C   64
#define GOUT_C   40
#define GALPHA   0.2f
#define GSLOPE   0.01f

// ---------------- elementwise / setup kernels ----------------

__global__ void k_fill(float* __restrict__ p, float val, int n) {
    int i = blockIdx.x * blockDim.x + threadIdx.x;
    if (i < n) p[i] = val;
}

__global__ void k_deg(const int* __restrict__ dst, float* __restrict__ deg, int E) {
    int e = blockIdx.x * blockDim.x + threadIdx.x;
    if (e < E) unsafeAtomicAdd(&deg[dst[e]], 1.0f);
}

__global__ void k_rsqrt(float* __restrict__ d, int n) {
    int i = blockIdx.x * blockDim.x + threadIdx.x;
    if (i < n) d[i] = rsqrtf(d[i]);   // deg >= 1 always (self loop)
}

__global__ void k_norm(const int* __restrict__ src, const int* __restrict__ dst,
                       const float* __restrict__ dinv, float* __restrict__ nrm, int E) {
    int e = blockIdx.x * blockDim.x + threadIdx.x;
    if (e < E) nrm[e] = dinv[src[e]] * dinv[dst[e]];
}

__global__ void k_pad_w4(const float* __restrict__ W4, const float* __restrict__ b4,
                         float* __restrict__ Wp, float* __restrict__ bp) {
    int i = blockIdx.x * blockDim.x + threadIdx.x;
    if (i < 48 * GHID_C) Wp[i] = (i < GOUT_C * GHID_C) ? W4[i] : 0.0f;
    if (i < 48)          bp[i] = (i < GOUT_C) ? b4[i] : 0.0f;
}

// ---------------- propagation: out = D^-1/2 (A+I) D^-1/2 in ----------------

// self-loop term initializes the output: out[i][:] = dinv[i]^2 * in[i][:]
__global__ void k_prop_self(const float* __restrict__ in, float* __restrict__ out,
                            const float* __restrict__ dinv, int n4, int Cq) {
    int i = blockIdx.x * blockDim.x + threadIdx.x;
    if (i >= n4) return;
    int node = i / Cq;
    float w = dinv[node]; w = w * w;
    float4 v = ((const float4*)in)[i];
    v.x *= w; v.y *= w; v.z *= w; v.w *= w;
    ((float4*)out)[i] = v;
}

// edge-parallel scatter-add: out[dst] += norm * in[src]   (float4 chunks)
__global__ void k_prop_scatter(const float* __restrict__ in, float* __restrict__ out,
                               const int* __restrict__ src, const int* __restrict__ dst,
                               const float* __restrict__ nrm, int E, int Cq, int C) {
    int t = blockIdx.x * blockDim.x + threadIdx.x;
    if (t >= E * Cq) return;
    int e = t / Cq, q = t % Cq;
    int s = src[e], d = dst[e];
    float w = nrm[e];
    float4 v = *(const float4*)(in + (size_t)s * C + q * 4);
    float* o = out + (size_t)d * C + q * 4;
    unsafeAtomicAdd(o + 0, w * v.x);
    unsafeAtomicAdd(o + 1, w * v.y);
    unsafeAtomicAdd(o + 2, w * v.z);
    unsafeAtomicAdd(o + 3, w * v.w);
}

// z = (1-alpha)*t + alpha*h0 ; optional leaky_relu
__global__ void k_combine(const float* __restrict__ t, const float* __restrict__ h0,
                          float* __restrict__ z, int n, int leaky) {
    int i = blockIdx.x * blockDim.x + threadIdx.x;
    if (i >= n) return;
    float v = (1.0f - GALPHA) * t[i] + GALPHA * h0[i];
    if (leaky) v = (v > 0.0f) ? v : GSLOPE * v;
    z[i] = v;
}

// ---------------- fp32 WMMA GEMM: out[M,ldc] = A[M,K] @ W[n,K]^T + bias ----------------
// one wave per 16x16 output tile; V_WMMA_F32_16X16X4_F32, K stepped by 4.
// A layout: lane m = lane&15, K pair selected by lane>>4 -> contiguous float2.
// B layout: lane n = lane&15, same K-pair selection -> contiguous float2 of W row n.
__global__ void k_gemm_wmma(const float* __restrict__ Ain, const float* __restrict__ W,
                            const float* __restrict__ bias, float* __restrict__ out,
                            int K, int ldc, int ncols) {
    const int lane  = threadIdx.x;        // 0..31
    const int ntile = threadIdx.y;
    const int half  = lane >> 4;          // 0 or 1
    const int lidx  = lane & 15;
    const int m = blockIdx.x * 16 + lidx; // A row for this lane
    const int n = ntile * 16 + lidx;      // B col (W row) for this lane
    const int kb = half * 2;              // K offset within 4-wide step

    const float* arow = Ain + (size_t)m * K + kb;
    const float* brow = W   + (size_t)n * K + kb;

    v8f acc = {};
    for (int k = 0; k < K; k += 4) {
        v2f a = *(const v2f*)(arow + k);
        v2f b = *(const v2f*)(brow + k);
        acc = __builtin_amdgcn_wmma_f32_16x16x4_f32(
            /*neg_a=*/false, a, /*neg_b=*/false, b,
            /*c_mod=*/(short)0, acc, /*reuse_a=*/false, /*reuse_b=*/false);
    }

    const int col = n;
    if (col < ncols) {
        float bv = bias[col];
        const int mbase = blockIdx.x * 16 + half * 8;
#pragma unroll
        for (int v = 0; v < 8; ++v)
            out[(size_t)(mbase + v) * ldc + col] = acc[v] + bv;
    }
}

// ---------------- log_softmax over rows of width C (<=64), wave per row ----------------

__global__ void k_logsoftmax(const float* __restrict__ in, float* __restrict__ out,
                             int nrows, int C) {
    const int lane = threadIdx.x & 31;
    const int row  = blockIdx.x * (blockDim.x >> 5) + (threadIdx.x >> 5);
    if (row >= nrows) return;
    const float* r = in + (size_t)row * C;
    float v0 = (lane      < C) ? r[lane]      : -INFINITY;
    float v1 = (lane + 32 < C) ? r[lane + 32] : -INFINITY;
    float m = fmaxf(v0, v1);
    for (int off = 16; off; off >>= 1) m = fmaxf(m, __shfl_xor(m, off, 32));
    float s = 0.0f;
    if (lane      < C) s += expf(v0 - m);
    if (lane + 32 < C) s += expf(v1 - m);
    for (int off = 16; off; off >>= 1) s += __shfl_xor(s, off, 32);
    float lse = m + logf(s);
    if (lane      < C) out[(size_t)row * C + lane]      = v0 - lse;
    if (lane + 32 < C) out[(size_t)row * C + lane + 32] = v1 - lse;
}

// ---------------- host orchestration ----------------

extern "C" void kernel_launch(void* const* d_in, const int* in_sizes, int n_in,
                              void* d_out, int out_size, void* d_ws, size_t ws_size,
                              hipStream_t stream) {
    const float* x  = (const float*)d_in[0];
    const int*   ei = (const int*)  d_in[1];
    const float* W0 = (const float*)d_in[2];
    const float* b0 = (const float*)d_in[3];
    const float* W4 = (const float*)d_in[4];
    const float* b4 = (const float*)d_in[5];

    const int N = in_sizes[0] / GIN_C;  // 50000
    const int E = in_sizes[1] / 2;      // 800000
    const int* src = ei;
    const int* dst = ei + E;

    float* ws = (float*)d_ws;
    size_t off = 0;
    float* dinv = ws + off; off += ((size_t)(N + 63) / 64) * 64;
    float* nrm  = ws + off; off += ((size_t)(E + 63) / 64) * 64;
    float* wpad = ws + off; off += 48 * GHID_C;
    float* bpad = ws + off; off += 64;
    float* buf0 = ws + off; off += (size_t)N * GIN_C;
    float* buf1 = ws + off; off += (size_t)N * GIN_C;
    float* buf2 = ws + off; off += (size_t)N * GHID_C;
    (void)ws_size; (void)n_in; (void)out_size;

    const int TB = 256;
    dim3 gN((N + TB - 1) / TB), gE((E + TB - 1) / TB);

    // degree -> dinv -> per-edge norm ; pad W4/b4 for the 48-wide GEMM2
    k_fill  <<<gN, TB, 0, stream>>>(dinv, 1.0f, N);
    k_deg   <<<gE, TB, 0, stream>>>(dst, dinv, E);
    k_rsqrt <<<gN, TB, 0, stream>>>(dinv, N);
    k_norm  <<<gE, TB, 0, stream>>>(src, dst, dinv, nrm, E);
    k_pad_w4<<<(48 * GHID_C + TB - 1) / TB, TB, 0, stream>>>(W4, b4, wpad, bpad);

    auto prop = [&](const float* pin, float* pout, int C) {
        int Cq = C / 4;
        int n4 = N * Cq;
        k_prop_self<<<(n4 + TB - 1) / TB, TB, 0, stream>>>(pin, pout, dinv, n4, Cq);
        long total = (long)E * Cq;
        k_prop_scatter<<<(int)((total + TB - 1) / TB), TB, 0, stream>>>(
            pin, pout, src, dst, nrm, E, Cq, C);
    };

    // conv0 (SGConv, K=2): prop^2 then linear 96 -> 64
    prop(x,    buf0, GIN_C);
    prop(buf0, buf1, GIN_C);
    dim3 gT(N / 16);
    k_gemm_wmma<<<gT, dim3(32, 4), 0, stream>>>(buf1, W0, b0, buf0, GIN_C, GHID_C, GHID_C);

    // 3x APPNP (K=2) + leaky_relu
    float* H  = buf0;   // h0 / current features
    float* Zb = buf1;   // z buffer
    const int NC = N * GHID_C;
    dim3 gNC((NC + TB - 1) / TB);
    for (int layer = 0; layer < 3; ++layer) {
        prop(H, buf2, GHID_C);
        k_combine<<<gNC, TB, 0, stream>>>(buf2, H, Zb, NC, 0);
        prop(Zb, buf2, GHID_C);
        k_combine<<<gNC, TB, 0, stream>>>(buf2, H, Zb, NC, 1);
        float* t = H; H = Zb; Zb = t;  // new features live in old Zb
    }

    // conv4 (SGConv, K=2): prop^2 then linear 64 -> 40 (padded to 48)
    prop(H,    buf2, GHID_C);
    prop(buf2, Zb,   GHID_C);
    k_gemm_wmma<<<gT, dim3(32, 3), 0, stream>>>(Zb, wpad, bpad, buf2, GHID_C, GOUT_C, GOUT_C);

    // log_softmax -> d_out
    const int wpb = 4;  // waves per block
    k_logsoftmax<<<(N + wpb - 1) / wpb, 32 * wpb, 0, stream>>>(buf2, (float*)d_out, N, GOUT_C);
}